// C2f_DualModal_MoE_52596169507002
// MI455X (gfx1250) — compile-verified
//
#include <hip/hip_runtime.h>

typedef __attribute__((ext_vector_type(2))) float v2f;
typedef __attribute__((ext_vector_type(8))) float v8f;

namespace cfg {
constexpr int B   = 8;
constexpr int C1  = 256;
constexpr int C2  = 256;
constexpr int HW  = 80 * 80;       // 6400
constexpr int E   = 4;             // routed experts
constexpr int S   = 1;             // shared experts
constexpr int KB  = 16;            // K block staged in LDS (double buffered)
constexpr int DT  = 64;            // d (C2) tile per workgroup
constexpr int NT  = 64;            // p (token) tile per workgroup
constexpr int XS_STR = 72;         // 2*72 % 64 == 16 -> half-waves hit disjoint bank groups
constexpr int AS_STR = 72;
}

#define WMMA4(Afrag, Bfrag, Cacc) \
    __builtin_amdgcn_wmma_f32_16x16x4_f32(false, (Afrag), false, (Bfrag), (short)0, (Cacc), false, false)

// Async DMA global -> LDS, 16B/lane, ASYNCcnt-tracked, optional 24-bit immediate offset.
#define ASYNC_G2L(lds_u32, gptr, imm)                                           \
    asm volatile("global_load_async_to_lds_b128 %0, %1, off offset:%c2"         \
                 :: "v"(lds_u32), "v"(gptr), "i"(imm) : "memory")
#define WAIT_ASYNC0() asm volatile("s_wait_asynccnt 0x0" ::: "memory")

// out[b,d,p] = sum_c x[b,c,p] * ( We[e*(b,p)][c][d] + mean_s Ws[s][c][d] )
// e*(b,p)    = argmax_e sum_c x[b,c,p] * Wr[c][e]   (top_k == 1 -> gate weight exactly 1.0)
__global__ __launch_bounds__(256)
void moe_1x1_wmma_f32(const float* __restrict__ x,
                      const float* __restrict__ Wr,
                      const float* __restrict__ We,
                      const float* __restrict__ Ws,
                      float* __restrict__ out)
{
    using namespace cfg;
    static_assert(S == 1, "async weight staging specializes shared-expert mean == Ws[0]");

    __shared__ __align__(16) float Xs[2][KB * XS_STR];      // X tile  [c][p], double buffered
    __shared__ __align__(16) float As[2][5 * KB * AS_STR];  // 4 experts + shared, [m][c][d]
    __shared__ __align__(16) float Wrs[C1 * E];             // full router weights
    __shared__ int eids[NT];                                // per-token expert id

    const int tid  = threadIdx.x;
    const int lane = tid & 31;
    const int lo   = lane & 15;    // M / column index within 16
    const int hi   = lane >> 4;    // K-pair selector (+2 rows for upper half-wave)
    const int wid  = tid >> 5;
    const int wd   = wid >> 1;     // wave's d sub-tile 0..3 (16 rows each)
    const int wn   = wid & 1;      // wave's p half 0..1 (32 cols each)

    // Provably wave-uniform router-wave flag -> scalar branch, no exec juggling.
    const bool rwave = (__builtin_amdgcn_readfirstlane(wd) == 0);

    const int pblk = blockIdx.x * NT;
    const int dblk = blockIdx.y * DT;
    const int b    = blockIdx.z;

    // Stage router weights once: 256 threads x float4 == 256 rows x 4 experts.
    ((float4*)Wrs)[tid] = ((const float4*)Wr)[tid];

    // ---- named accumulators (kept strictly in VGPRs; no arrays -> no alloca) ----
    const v8f vz = {0.f, 0.f, 0.f, 0.f, 0.f, 0.f, 0.f, 0.f};
    v8f A00 = vz, A01 = vz, A02 = vz, A03 = vz, A04 = vz;   // p-subtile 0: experts 0..3, shared
    v8f A10 = vz, A11 = vz, A12 = vz, A13 = vz, A14 = vz;   // p-subtile 1
    v8f R0 = vz, R1 = vz;                                   // router logits (rwave only)

    // ---- loop-carried DMA source pointers (constant stride per K block) ----
    const int cq = tid >> 4;            // 0..15: K row within block
    const int q4 = (tid & 15) << 2;     // 0..60: 4-aligned column
    const float* xptr = x  + (size_t)b * C1 * HW + (size_t)cq * HW + pblk + q4;
    const float* wptr = We + (size_t)cq * C2 + dblk + q4;   // +mm*C1*C2 via offset: imm
    const float* sptr = Ws + (size_t)cq * C2 + dblk + q4;   // S==1: mean == Ws[0]

    // ---- per-thread LDS destinations (hoisted once) ----
    const unsigned dX0  = (unsigned)(uintptr_t)&Xs[0][cq * XS_STR + q4];
    const unsigned dX1  = (unsigned)(uintptr_t)&Xs[1][cq * XS_STR + q4];
    const unsigned dA00 = (unsigned)(uintptr_t)&As[0][(0 * KB + cq) * AS_STR + q4];
    const unsigned dA01 = (unsigned)(uintptr_t)&As[0][(1 * KB + cq) * AS_STR + q4];
    const unsigned dA02 = (unsigned)(uintptr_t)&As[0][(2 * KB + cq) * AS_STR + q4];
    const unsigned dA03 = (unsigned)(uintptr_t)&As[0][(3 * KB + cq) * AS_STR + q4];
    const unsigned dA04 = (unsigned)(uintptr_t)&As[0][(4 * KB + cq) * AS_STR + q4];
    const unsigned dA10 = (unsigned)(uintptr_t)&As[1][(0 * KB + cq) * AS_STR + q4];
    const unsigned dA11 = (unsigned)(uintptr_t)&As[1][(1 * KB + cq) * AS_STR + q4];
    const unsigned dA12 = (unsigned)(uintptr_t)&As[1][(2 * KB + cq) * AS_STR + q4];
    const unsigned dA13 = (unsigned)(uintptr_t)&As[1][(3 * KB + cq) * AS_STR + q4];
    const unsigned dA14 = (unsigned)(uintptr_t)&As[1][(4 * KB + cq) * AS_STR + q4];

#define ISSUE(BUF, PF)                                                          \
    do {                                                                        \
        ASYNC_G2L(dX##BUF, xptr, 0);                                            \
        ASYNC_G2L(dA##BUF##0, wptr, 0 * C1 * C2 * 4);                           \
        ASYNC_G2L(dA##BUF##1, wptr, 1 * C1 * C2 * 4);                           \
        ASYNC_G2L(dA##BUF##2, wptr, 2 * C1 * C2 * 4);                           \
        ASYNC_G2L(dA##BUF##3, wptr, 3 * C1 * C2 * 4);                           \
        ASYNC_G2L(dA##BUF##4, sptr, 0);                                         \
        if (PF) __builtin_prefetch(xptr + (size_t)KB * HW, 0, 3);               \
        xptr += (size_t)KB * HW;                                                \
        wptr += (size_t)KB * C2;                                                \
        sptr += (size_t)KB * C2;                                                \
    } while (0)

#define COMPUTE_PHASE(BUF, KBASE)                                               \
    do {                                                                        \
        _Pragma("unroll")                                                       \
        for (int st = 0; st < KB / 4; ++st) {                                   \
            const int c0   = 4 * st + 2 * hi;                                   \
            const int colb = 32 * wn + lo;                                      \
            const v2f bf0 = {Xs[BUF][c0 * XS_STR + colb],                       \
                             Xs[BUF][(c0 + 1) * XS_STR + colb]};                \
            const v2f bf1 = {Xs[BUF][c0 * XS_STR + colb + 16],                  \
                             Xs[BUF][(c0 + 1) * XS_STR + colb + 16]};           \
            const int ab  = c0 * AS_STR + 16 * wd + lo;                         \
            const v2f af0 = {As[BUF][ab + 0 * KB * AS_STR],                     \
                             As[BUF][ab + 0 * KB * AS_STR + AS_STR]};           \
            const v2f af1 = {As[BUF][ab + 1 * KB * AS_STR],                     \
                             As[BUF][ab + 1 * KB * AS_STR + AS_STR]};           \
            const v2f af2 = {As[BUF][ab + 2 * KB * AS_STR],                     \
                             As[BUF][ab + 2 * KB * AS_STR + AS_STR]};           \
            const v2f af3 = {As[BUF][ab + 3 * KB * AS_STR],                     \
                             As[BUF][ab + 3 * KB * AS_STR + AS_STR]};           \
            const v2f af4 = {As[BUF][ab + 4 * KB * AS_STR],                     \
                             As[BUF][ab + 4 * KB * AS_STR + AS_STR]};           \
            A00 = WMMA4(af0, bf0, A00);  A10 = WMMA4(af0, bf1, A10);            \
            A01 = WMMA4(af1, bf0, A01);  A11 = WMMA4(af1, bf1, A11);            \
            A02 = WMMA4(af2, bf0, A02);  A12 = WMMA4(af2, bf1, A12);            \
            A03 = WMMA4(af3, bf0, A03);  A13 = WMMA4(af3, bf1, A13);            \
            A04 = WMMA4(af4, bf0, A04);  A14 = WMMA4(af4, bf1, A14);            \
            if (rwave) { /* scalar branch; EXEC all-ones for the WMMAs */       \
                const int cg = (KBASE) + c0;                                    \
                const float w0 = (lo < E) ? Wrs[cg * E + lo] : 0.0f;            \
                const float w1 = (lo < E) ? Wrs[(cg + 1) * E + lo] : 0.0f;      \
                const v2f ar = {w0, w1};                                        \
                R0 = WMMA4(ar, bf0, R0);                                        \
                R1 = WMMA4(ar, bf1, R1);                                        \
            }                                                                   \
        }                                                                       \
    } while (0)

    // ---- software pipeline: DMA block k+1 overlaps the 48 WMMAs of block k ----
    ISSUE(0, true);                 // prologue: block 0 -> buf0
    WAIT_ASYNC0();
    __syncthreads();                // publishes buf0 (and Wrs)

    for (int kb = 0; kb < C1; kb += 2 * KB) {       // 8 iterations, 2 blocks each
        // phase A: compute buf0 (block kb), DMA block kb+KB -> buf1
        ISSUE(1, kb + 2 * KB < C1);                 // kb+KB < C1 always holds here
        COMPUTE_PHASE(0, kb);
        WAIT_ASYNC0();
        __syncthreads();

        // phase B: compute buf1 (block kb+KB), DMA block kb+2*KB -> buf0
        if (kb + 2 * KB < C1) ISSUE(0, kb + 3 * KB < C1);
        COMPUTE_PHASE(1, kb + KB);
        WAIT_ASYNC0();
        __syncthreads();
    }
#undef ISSUE
#undef COMPUTE_PHASE

    // ---- router argmax: logits of token-column j sit in lane j, elements 0..3 ----
    if (rwave) {
        if (lane < 16) {
            float best = R0[0]; int e = 0;
            if (R0[1] > best) { best = R0[1]; e = 1; }
            if (R0[2] > best) { best = R0[2]; e = 2; }
            if (R0[3] > best) { best = R0[3]; e = 3; }
            eids[32 * wn + lane] = e;               // strict '>' == top_k lowest-index tiebreak

            best = R1[0]; e = 0;
            if (R1[1] > best) { best = R1[1]; e = 1; }
            if (R1[2] > best) { best = R1[2]; e = 2; }
            if (R1[3] > best) { best = R1[3]; e = 3; }
            eids[32 * wn + 16 + lane] = e;
        }
    }
    __syncthreads();

    // ---- gate select + shared add + store (rows contiguous in p -> coalesced) ----
    {
        const int d0 = dblk + 16 * wd + 8 * hi;     // vector element v holds row d0 + v
        float* ob = out + ((size_t)b * C2 + d0) * HW + pblk;

        const int e0 = eids[32 * wn + lo];
        v8f sel = A00;
        sel = (e0 == 1) ? A01 : sel;
        sel = (e0 == 2) ? A02 : sel;
        sel = (e0 == 3) ? A03 : sel;
        const v8f o0 = sel + A04;
        float* op0 = ob + 32 * wn + lo;
#pragma unroll
        for (int v = 0; v < 8; ++v) op0[(size_t)v * HW] = o0[v];

        const int e1 = eids[32 * wn + 16 + lo];
        sel = A10;
        sel = (e1 == 1) ? A11 : sel;
        sel = (e1 == 2) ? A12 : sel;
        sel = (e1 == 3) ? A13 : sel;
        const v8f o1 = sel + A14;
        float* op1 = ob + 32 * wn + 16 + lo;
#pragma unroll
        for (int v = 0; v < 8; ++v) op1[(size_t)v * HW] = o1[v];
    }
}

extern "C" void kernel_launch(void* const* d_in, const int* in_sizes, int n_in,
                              void* d_out, int out_size, void* d_ws, size_t ws_size,
                              hipStream_t stream) {
    using namespace cfg;
    const float* x  = (const float*)d_in[0];
    const float* Wr = (const float*)d_in[1];
    const float* We = (const float*)d_in[2];
    const float* Ws = (const float*)d_in[3];
    // d_in[4] is top_k == 1 in this configuration; kernel specializes top-1.
    float* out = (float*)d_out;

    dim3 grid(HW / NT, C2 / DT, B);   // 100 x 4 x 8 workgroups
    moe_1x1_wmma_f32<<<grid, dim3(256), 0, stream>>>(x, Wr, We, Ws, out);
}